// mpNet_76287209112059
// MI455X (gfx1250) — compile-verified
//
#include <hip/hip_runtime.h>

// mpNet: complex matching pursuit on MI455X (gfx1250).
// - f32 WMMA (V_WMMA_F32_16X16X4_F32) for the [16,64]x[64,16] complex score tiles
// - residual kept entirely in registers in WMMA A-fragment layout
// - W tiles staged block-wide into LDS via async DMA (global_load_async_to_lds_b128,
//   ASYNCcnt / s_wait_asynccnt), double-buffered.

typedef float v2f __attribute__((ext_vector_type(2)));
typedef float v8f __attribute__((ext_vector_type(8)));

#define N_SAMPLES 8192
#define M_DIM 64
#define A_DIM 2048
#define ROWS_PER_WAVE 16
#define WAVES_PER_BLOCK 4
#define THREADS_PER_BLOCK (32 * WAVES_PER_BLOCK)
#define NT_TILES (A_DIM / 16)     // 128 column tiles
#define KSTEPS (M_DIM / 4)        // 16 k-steps of V_WMMA_F32_16X16X4_F32
#define TILE_FLOATS (M_DIM * 16)  // 1024 floats = 4 KB per matrix per tile

// Async DMA: each lane copies 16 bytes from global address gsrc to LDS address ldst, tracked by ASYNCcnt.
__device__ __forceinline__ void async_copy_b128(const float* gsrc, const float* ldst) {
    // Generic pointer to LDS: low 32 bits are the LDS byte address (ISA 10.2).
    asm volatile("global_load_async_to_lds_b128 %0, %1, off"
                 :: "v"((unsigned int)(unsigned long long)ldst), "v"(gsrc)
                 : "memory");
}
__device__ __forceinline__ void wait_async0() {
    asm volatile("s_wait_asynccnt 0x0" ::: "memory");
}

__global__ __launch_bounds__(THREADS_PER_BLOCK)
void mp_wmma_kernel(const float* __restrict__ x_re,
                    const float* __restrict__ x_im,
                    const float* __restrict__ w_re,
                    const float* __restrict__ w_im,
                    const int*   __restrict__ kptr,
                    float*       __restrict__ out)
{
    const int tid  = threadIdx.x;
    const int lane = tid & 31;
    const int wib  = tid >> 5;                               // wave in block
    const int waveGlobal = blockIdx.x * WAVES_PER_BLOCK + wib;
    const int row0 = waveGlobal * ROWS_PER_WAVE;
    const int r  = lane & 15;                                // tile row (M) / tile col (N)
    const int hi = lane >> 4;                                // lane half selects K pair
    const int row = row0 + r;
    const int kiters = *kptr;

    // Double-buffered W tile: [buf][re/im][k*16 + n], row-major 64x16.
    __shared__ float wtile[2][2][TILE_FLOATS];
    // Cross-lane argmax scratch (per wave region; all barriers are block-uniform).
    __shared__ float red_v2[WAVES_PER_BLOCK][32][8];
    __shared__ float red_re[WAVES_PER_BLOCK][32][8];
    __shared__ float red_im[WAVES_PER_BLOCK][32][8];
    __shared__ int   red_ix[WAVES_PER_BLOCK][32][8];
    __shared__ float row_re[WAVES_PER_BLOCK][16];
    __shared__ float row_im[WAVES_PER_BLOCK][16];
    __shared__ int   row_ix[WAVES_PER_BLOCK][16];

    // Pull W toward L2 / WGP$ early (global_prefetch_b8).
    __builtin_prefetch((const void*)(w_re + ((size_t)tid << 9)), 0, 1);
    __builtin_prefetch((const void*)(w_im + ((size_t)tid << 9)), 0, 1);

    // Residual in registers, A-fragment layout of V_WMMA_F32_16X16X4_F32:
    // lane l holds row l&15, K = 4*ks + 2*(l>>4) + {0,1}.
    v2f ar[KSTEPS], ai[KSTEPS];
#pragma unroll
    for (int ks = 0; ks < KSTEPS; ++ks) {
        const int m0 = 4 * ks + 2 * hi;
        const int base = row * M_DIM + m0;                   // 8B aligned -> b64 loads
        ar[ks] = *(const v2f*)(x_re + base);
        ai[ks] = *(const v2f*)(x_im + base);
    }

    for (int it = 0; it < kiters; ++it) {
        // --- per-lane running argmax over this lane's column class ---
        float bv[8], bre[8], bim[8]; int bix[8];
#pragma unroll
        for (int v = 0; v < 8; ++v) { bv[v] = -1.0f; bre[v] = 0.0f; bim[v] = 0.0f; bix[v] = 0; }

        // Prologue: stage tile 0 into buffer 0 (2 b128 chunks per thread per matrix).
        {
#pragma unroll
            for (int j = 0; j < 2; ++j) {
                const int c = tid + j * 128;                 // chunk 0..255
                const int k = c >> 2, cc = c & 3;
                async_copy_b128(w_re + k * A_DIM + cc * 4, &wtile[0][0][c * 4]);
                async_copy_b128(w_im + k * A_DIM + cc * 4, &wtile[0][1][c * 4]);
            }
        }

        for (int nt = 0; nt < NT_TILES; ++nt) {
            const int cur = nt & 1;
            wait_async0();        // this wave's DMAs for tile nt are complete
            __syncthreads();      // everyone's tile-nt DMAs visible; buf[cur^1] fully consumed
            if (nt + 1 < NT_TILES) {
                const int n1 = (nt + 1) * 16;
#pragma unroll
                for (int j = 0; j < 2; ++j) {
                    const int c = tid + j * 128;
                    const int k = c >> 2, cc = c & 3;
                    async_copy_b128(w_re + k * A_DIM + n1 + cc * 4, &wtile[cur ^ 1][0][c * 4]);
                    async_copy_b128(w_im + k * A_DIM + n1 + cc * 4, &wtile[cur ^ 1][1][c * 4]);
                }
            }

            v8f cre = {};   // s_re accumulator: r_re@W_re + r_im@W_im  (conj(W))
            v8f cb  = {};   // r_im@W_re
            v8f cc2 = {};   // r_re@W_im  (s_im = cb - cc2)
            const float* tre = &wtile[cur][0][0];
            const float* tim = &wtile[cur][1][0];
#pragma unroll
            for (int ks = 0; ks < KSTEPS; ++ks) {
                // B fragment: V0 = K = 4*ks+2*hi, V1 = K = 4*ks+2*hi+1, col r (bank-conflict-free)
                const int rb = (4 * ks + 2 * hi) * 16 + r;
                v2f br, bi;
                br.x = tre[rb]; br.y = tre[rb + 16];
                bi.x = tim[rb]; bi.y = tim[rb + 16];
                cre = __builtin_amdgcn_wmma_f32_16x16x4_f32(false, ar[ks], false, br, (short)0, cre, false, false);
                cre = __builtin_amdgcn_wmma_f32_16x16x4_f32(false, ai[ks], false, bi, (short)0, cre, false, false);
                cb  = __builtin_amdgcn_wmma_f32_16x16x4_f32(false, ai[ks], false, br, (short)0, cb,  false, false);
                cc2 = __builtin_amdgcn_wmma_f32_16x16x4_f32(false, ar[ks], false, bi, (short)0, cc2, false, false);
            }
            const int n = nt * 16 + r;                       // N column this lane holds
#pragma unroll
            for (int v = 0; v < 8; ++v) {                    // rows M = v + 8*hi
                const float sre = cre[v];
                const float sim = cb[v] - cc2[v];
                const float m2 = sre * sre + sim * sim;
                if (m2 > bv[v] || (m2 == bv[v] && n < bix[v])) {
                    bv[v] = m2; bix[v] = n; bre[v] = sre; bim[v] = sim;
                }
            }
        }

        // --- cross-lane reduction: per row, max over its 16 column-class lanes ---
#pragma unroll
        for (int v = 0; v < 8; ++v) {
            red_v2[wib][lane][v] = bv[v];
            red_re[wib][lane][v] = bre[v];
            red_im[wib][lane][v] = bim[v];
            red_ix[wib][lane][v] = bix[v];
        }
        __syncthreads();
        if (lane < 16) {
            // lane == row in tile. Row rr<8 lives in lanes 0..15 slot rr; rr>=8 in lanes 16..31 slot rr-8.
            const int srcBase = (lane < 8) ? 0 : 16;
            const int slot = lane & 7;
            float bestv = -1.0f, bsre = 0.0f, bsim = 0.0f; int bidx = 0x7fffffff;
            for (int sl = 0; sl < 16; ++sl) {
                const int s = srcBase + sl;
                const float v2 = red_v2[wib][s][slot];
                const int   ix = red_ix[wib][s][slot];
                if (v2 > bestv || (v2 == bestv && ix < bidx)) {
                    bestv = v2; bidx = ix;
                    bsre = red_re[wib][s][slot];
                    bsim = red_im[wib][s][slot];
                }
            }
            row_re[wib][lane] = bsre;
            row_im[wib][lane] = bsim;
            row_ix[wib][lane] = bidx;
        }
        __syncthreads();

        // --- residual update: residual[row] -= val * W[:, idx]  (no conj) ---
        const float vre = row_re[wib][r];
        const float vim = row_im[wib][r];
        const int   idx = row_ix[wib][r];
        const float* wtr = w_re + idx;
        const float* wti = w_im + idx;
#pragma unroll
        for (int ks = 0; ks < KSTEPS; ++ks) {
            const int m0 = 4 * ks + 2 * hi;
            const float wr0 = wtr[m0 * A_DIM], wr1 = wtr[(m0 + 1) * A_DIM];
            const float wi0 = wti[m0 * A_DIM], wi1 = wti[(m0 + 1) * A_DIM];
            ar[ks].x -= vre * wr0 - vim * wi0;
            ar[ks].y -= vre * wr1 - vim * wi1;
            ai[ks].x -= vre * wi0 + vim * wr0;
            ai[ks].y -= vre * wi1 + vim * wr1;
        }
        // Next iteration's LDS writes are ordered by its own wait+barrier protocol.
    }

    // --- outputs: residual (complex interleaved), then x_hat = x - residual ---
    const size_t NM2 = (size_t)N_SAMPLES * M_DIM * 2;
#pragma unroll
    for (int ks = 0; ks < KSTEPS; ++ks) {
        const int m0 = 4 * ks + 2 * hi;
        const size_t base = (size_t)row * M_DIM + m0;
        out[2 * base + 0] = ar[ks].x;
        out[2 * base + 1] = ai[ks].x;
        out[2 * base + 2] = ar[ks].y;
        out[2 * base + 3] = ai[ks].y;
        const float xr0 = x_re[base], xr1 = x_re[base + 1];
        const float xi0 = x_im[base], xi1 = x_im[base + 1];
        out[NM2 + 2 * base + 0] = xr0 - ar[ks].x;
        out[NM2 + 2 * base + 1] = xi0 - ai[ks].x;
        out[NM2 + 2 * base + 2] = xr1 - ar[ks].y;
        out[NM2 + 2 * base + 3] = xi1 - ai[ks].y;
    }
}

extern "C" void kernel_launch(void* const* d_in, const int* in_sizes, int n_in,
                              void* d_out, int out_size, void* d_ws, size_t ws_size,
                              hipStream_t stream) {
    (void)in_sizes; (void)n_in; (void)out_size; (void)d_ws; (void)ws_size;
    const float* x_re = (const float*)d_in[0];
    const float* x_im = (const float*)d_in[1];
    const float* w_re = (const float*)d_in[2];
    const float* w_im = (const float*)d_in[3];
    const int*   kptr = (const int*)d_in[4];
    float* out = (float*)d_out;

    const int blocks = N_SAMPLES / (ROWS_PER_WAVE * WAVES_PER_BLOCK); // 128
    mp_wmma_kernel<<<blocks, THREADS_PER_BLOCK, 0, stream>>>(x_re, x_im, w_re, w_im, kptr, out);
}